// HopfRNNCellTheta_80221399154876
// MI455X (gfx1250) — compile-verified
//
#include <hip/hip_runtime.h>

typedef __attribute__((ext_vector_type(2))) float v2f;
typedef __attribute__((ext_vector_type(4))) float v4f;
typedef __attribute__((ext_vector_type(8))) float v8f;

static constexpr int Ndim  = 1024;
static constexpr int UNITS = 1024;
static constexpr int Tdim  = 64;

// ---------------------------------------------------------------------------
// Kernel 1: complex GEMM  Y[n,t] = 0.5 * sum_m B[n,m] * X[m,t]
// X[m,t] = x[t*N + m]  (x stored (T,N,1) row-major)
// One wave (32 lanes) owns one 16x16 tile of Y; K swept in steps of 4 with
// V_WMMA_F32_16X16X4_F32 (pure fp32, matches reference precision).
//
// A-operand (16x4 f32) lane layout: lanes 0-15 hold K={0,1} in v[0],v[1],
// lanes 16-31 hold K={2,3}.  B-operand (4x16) mirrors it with N striped
// across lanes.  Both become contiguous 8-byte loads along K.
// ---------------------------------------------------------------------------
__global__ __launch_bounds__(32)
void hopf_gemm(const float* __restrict__ xre, const float* __restrict__ xim,
               const float* __restrict__ Bre, const float* __restrict__ Bim,
               float2* __restrict__ Yc)
{
    const int tile = blockIdx.x;            // 0..255
    const int n0   = (tile & 63) * 16;      // 64 row tiles
    const int t0   = (tile >> 6) * 16;      // 4  col tiles
    const int lane = threadIdx.x;           // 0..31 (wave32)
    const int half = lane >> 4;             // selects K pair {0,1} vs {2,3}
    const int sub  = lane & 15;             // row (A) / col (B) within tile

    const float* aRe = Bre + (size_t)(n0 + sub) * Ndim + 2 * half;
    const float* aIm = Bim + (size_t)(n0 + sub) * Ndim + 2 * half;
    const float* bRe = xre + (size_t)(t0 + sub) * Ndim + 2 * half;  // X[:,t] contiguous
    const float* bIm = xim + (size_t)(t0 + sub) * Ndim + 2 * half;

    v8f rr = {}, ii = {}, ri = {}, ir = {};

#pragma unroll 4
    for (int k = 0; k < Ndim; k += 4) {
        v2f ar = *(const v2f*)(aRe + k);
        v2f ai = *(const v2f*)(aIm + k);
        v2f br = *(const v2f*)(bRe + k);
        v2f bi = *(const v2f*)(bIm + k);
        rr = __builtin_amdgcn_wmma_f32_16x16x4_f32(false, ar, false, br, (short)0, rr, false, false);
        ii = __builtin_amdgcn_wmma_f32_16x16x4_f32(false, ai, false, bi, (short)0, ii, false, false);
        ri = __builtin_amdgcn_wmma_f32_16x16x4_f32(false, ar, false, bi, (short)0, ri, false, false);
        ir = __builtin_amdgcn_wmma_f32_16x16x4_f32(false, ai, false, br, (short)0, ir, false, false);
    }

    // C/D layout: vgpr r, lanes 0-15 -> m=r, lanes 16-31 -> m=8+r; col = sub.
#pragma unroll
    for (int r = 0; r < 8; ++r) {
        const int m = r + 8 * half;
        const float yre = 0.5f * (rr[r] - ii[r]);
        const float yim = 0.5f * (ri[r] + ir[r]);
        Yc[(size_t)(n0 + m) * Tdim + (t0 + sub)] = make_float2(yre, yim);
    }
}

// ---------------------------------------------------------------------------
// Kernel 2: streaming writer (the real cost: 537 MB of stores).
// ys[t, n, u] = 0                       if u <  UNITS-(t+1)
//             = Y[n, u - (UNITS-t-1)]   otherwise
// One block per (t,n) row (8 KB). Each thread owns 4 consecutive complex
// values -> two 16B non-temporal stores (32 contiguous bytes per thread).
// ---------------------------------------------------------------------------
__global__ __launch_bounds__(256)
void hopf_write(const float2* __restrict__ Yc, float2* __restrict__ out)
{
    const int row = blockIdx.x;           // row = t*N + n
    const int t   = row >> 10;            // / Ndim
    const int n   = row & (Ndim - 1);
    const int u0  = UNITS - (t + 1);      // first non-zero column

    float2*       orow = out + (size_t)row * UNITS;
    const float2* yrow = Yc + (size_t)n * Tdim;

    const int u = threadIdx.x * 4;

    float2 v[4];
#pragma unroll
    for (int j = 0; j < 4; ++j) {
        const int uu = u + j;
        v[j] = (uu >= u0) ? yrow[uu - u0] : make_float2(0.0f, 0.0f);
    }

    v4f* o4 = (v4f*)(orow + u);           // 32B-aligned (u multiple of 4)
    v4f lo = { v[0].x, v[0].y, v[1].x, v[1].y };
    v4f hi = { v[2].x, v[2].y, v[3].x, v[3].y };
    __builtin_nontemporal_store(lo, o4 + 0);
    __builtin_nontemporal_store(hi, o4 + 1);
}

// ---------------------------------------------------------------------------
extern "C" void kernel_launch(void* const* d_in, const int* in_sizes, int n_in,
                              void* d_out, int out_size, void* d_ws, size_t ws_size,
                              hipStream_t stream)
{
    (void)in_sizes; (void)n_in; (void)out_size; (void)ws_size;

    const float* xre = (const float*)d_in[0];   // (T, N, 1)
    const float* xim = (const float*)d_in[1];
    // d_in[2], d_in[3] = A_re, A_im: dead (only ever multiply the all-zero
    // leading column of z, since T < UNITS).
    const float* Bre = (const float*)d_in[4];   // (N, N)
    const float* Bim = (const float*)d_in[5];

    float2* Yc = (float2*)d_ws;                 // N*T complex = 512 KB scratch

    hopf_gemm<<<(Ndim / 16) * (Tdim / 16), 32, 0, stream>>>(xre, xim, Bre, Bim, Yc);
    hopf_write<<<Tdim * Ndim, 256, 0, stream>>>(Yc, (float2*)d_out);
}